// BaseSpanProposer_73650099192304
// MI455X (gfx1250) — compile-verified
//
#include <hip/hip_runtime.h>
#include <hip/hip_bf16.h>
#include <stdint.h>

#define B_ 128
#define S_ 512
#define L_ 20
#define N_ (B_ * S_ * L_)   // 1310720 candidate slots

// Output layout (floats), concatenated in reference return order:
//  [0      ,  1*N_) viable
//  [1*N_   ,  2*N_) viable_batch_indices
//  [2*N_   ,  3*N_) viable_starts
//  [3*N_   ,  4*N_) viable_ends
//  [4*N_   ,  5*N_) viable_lengths
//  [5*N_   , 25*N_) viable_unit_id_seqs (N_ x L_)
//  [25*N_  , 26*N_) viable_p_weights
//  [26*N_  , 27*N_) start_candidates
//  [27*N_  , 28*N_) end_candidates
//  [28*N_  , 29*N_) len_candidates

// meta (in d_ws, int32): meta[0..127] = exclusive per-batch prefix of viable
// counts; meta[128] = total viable count; meta[129] = max(lost_lengths).

// Closed-form sum_{s'=0}^{s-1} min(L, max(0, M-s'))
__device__ __forceinline__ int q_prefix(int s, int M) {
    int t = min(s, M - L_ + 1);     // rows contributing L each
    int u = min(s, M);              // end of partial region
    int k = u - t;                  // partial rows
    return L_ * t + k * M - (t + u - 1) * k / 2;  // product always even
}

__global__ void span_setup(const int* __restrict__ lens, int* __restrict__ meta) {
    __shared__ int cb[B_];
    int b = threadIdx.x;
    int M = lens[b];
    cb[b] = q_prefix(S_, M);        // C_b = Q(S)
    __syncthreads();
    if (b == 0) {
        int acc = 0, mx = 0;
        for (int i = 0; i < B_; ++i) { meta[i] = acc; acc += cb[i]; }
        for (int i = 0; i < B_; ++i) mx = max(mx, lens[i]);
        meta[B_]     = acc;   // N_total
        meta[B_ + 1] = mx;    // max_length
    }
}

// Broadcast outputs: viable + start/end/len candidates, 4 flat indices per
// thread, b128 stores. Group of 4 never straddles an s (20) or b (10240)
// boundary since both are multiples of 4.
__global__ __launch_bounds__(256) void span_cand(const int* __restrict__ lens,
                                                 float* __restrict__ out) {
    int g  = blockIdx.x * 256 + threadIdx.x;   // grid covers N_/4 groups
    int i0 = g * 4;
    int l0 = i0 % L_;                          // in {0,4,8,12,16}
    int sb = i0 / L_;
    int s  = sb & (S_ - 1);
    int b  = sb >> 9;
    int M  = lens[b];

    float4 vi, en, ln;
    int e0 = s + l0;
    vi.x = (e0     < M) ? 1.0f : 0.0f;
    vi.y = (e0 + 1 < M) ? 1.0f : 0.0f;
    vi.z = (e0 + 2 < M) ? 1.0f : 0.0f;
    vi.w = (e0 + 3 < M) ? 1.0f : 0.0f;
    float st = (float)s;
    float4 stv = make_float4(st, st, st, st);
    en = make_float4((float)e0, (float)(e0 + 1), (float)(e0 + 2), (float)(e0 + 3));
    ln = make_float4((float)(l0 + 1), (float)(l0 + 2), (float)(l0 + 3), (float)(l0 + 4));

    size_t base = (size_t)26 * N_;
    *(float4*)(out + i0)                         = vi;   // viable
    *(float4*)(out + base + i0)                  = stv;  // start_candidates
    *(float4*)(out + base + (size_t)N_ + i0)     = en;   // end_candidates
    *(float4*)(out + base + (size_t)2 * N_ + i0) = ln;   // len_candidates
}

// One wave32 per (b,s) row: analytic compaction offset; unit-id row staged
// once in LDS via CDNA5 async global->LDS load, converted to float once,
// replicated c times with b128 stores.
__global__ __launch_bounds__(256) void span_compact(
    const int* __restrict__ seqs, const int* __restrict__ lens,
    const float* __restrict__ pw, const int* __restrict__ meta,
    float* __restrict__ out) {
    __shared__ int   rowbuf[8][32];
    __shared__ float fbuf[8][20];     // 80B row pitch -> 16B aligned rows
    const int warp = threadIdx.x >> 5;
    const int lane = threadIdx.x & 31;
    const int row  = blockIdx.x * 8 + warp;   // 0 .. B_*S_-1
    const int b    = row >> 9;
    const int s    = row & (S_ - 1);
    const int M      = lens[b];
    const int maxlen = meta[B_ + 1];

    // Async gather of the (clamped) unit row into LDS; ASYNCcnt tracks it.
    const int pos = min(s + lane, maxlen - 1);
    const int* gp = seqs + b * S_ + pos;
    uint32_t ldsoff = (uint32_t)(uint64_t)(&rowbuf[warp][lane]);
    asm volatile("global_load_async_to_lds_b32 %0, %1, off"
                 :: "v"(ldsoff), "v"((uint64_t)(uintptr_t)gp) : "memory");

    const int c = min(L_, max(0, M - s));
    const int o = meta[b] + q_prefix(s, M);   // compacted base offset

    if (lane < c) {
        int j = o + lane;
        out[(size_t)1 * N_ + j]  = (float)b;            // batch index
        out[(size_t)2 * N_ + j]  = (float)s;            // start
        out[(size_t)3 * N_ + j]  = (float)(s + lane);   // end
        out[(size_t)4 * N_ + j]  = (float)(lane + 1);   // length
        out[(size_t)25 * N_ + j] = pw[(size_t)row * L_ + lane];
    }

    asm volatile("s_wait_asynccnt 0x0" ::: "memory");

    // Convert int ids -> float once (LDS ops of a wave are in-order).
    if (lane < L_) fbuf[warp][lane] = (float)rowbuf[warp][lane];

    // Write c identical L_-wide rows as contiguous b128 stores.
    // Region [o*20, (o+c)*20) floats: 16B aligned, multiple of 4 floats.
    const int tot4 = c * (L_ / 4);                       // float4 count
    float4* __restrict__ dst = (float4*)(out + (size_t)5 * N_ + (size_t)o * L_);
    const float4* fv = (const float4*)fbuf[warp];
    for (int f = lane; f < tot4; f += 32)
        dst[f] = fv[f % 5];
}

// Fill [N_total, N_) of the compacted streams with nonzero's fill_value=0
// replica: flat index 0 => b=0, s=0, l=0 (end=0, len=1, pw[0], seqs[0,0:20]).
__global__ __launch_bounds__(256) void span_tail(const int* __restrict__ seqs,
                                                 const float* __restrict__ pw,
                                                 const int* __restrict__ meta,
                                                 float* __restrict__ out) {
    __shared__ float4 rows0v[5];
    __shared__ float  pw0s;
    if (threadIdx.x < 5) {
        int t = threadIdx.x * 4;
        rows0v[threadIdx.x] = make_float4((float)seqs[t], (float)seqs[t + 1],
                                          (float)seqs[t + 2], (float)seqs[t + 3]);
    }
    if (threadIdx.x == 0) pw0s = pw[0];
    __syncthreads();

    const int Ntot = meta[B_];
    const int base = blockIdx.x * 256;        // grid covers exactly N_ rows
    const int i    = base + threadIdx.x;

    if (i >= Ntot) {                          // scalar compacted streams
        out[(size_t)1 * N_ + i]  = 0.0f;
        out[(size_t)2 * N_ + i]  = 0.0f;
        out[(size_t)3 * N_ + i]  = 0.0f;
        out[(size_t)4 * N_ + i]  = 1.0f;
        out[(size_t)25 * N_ + i] = pw0s;
    }

    // Unit rows: block-contiguous b128 stores over [lo_row*20, hi_row*20).
    const int lo_row = max(Ntot, base);
    const int hi_row = base + 256;
    if (lo_row < hi_row) {
        int u_lo = lo_row * (L_ / 4);         // float4 index (16B aligned)
        int u_hi = hi_row * (L_ / 4);
        float4* __restrict__ ud = (float4*)(out + (size_t)5 * N_);
        for (int f = u_lo + threadIdx.x; f < u_hi; f += 256)
            ud[f] = rows0v[f % 5];
    }
}

extern "C" void kernel_launch(void* const* d_in, const int* in_sizes, int n_in,
                              void* d_out, int out_size, void* d_ws, size_t ws_size,
                              hipStream_t stream) {
    const int*   seqs = (const int*)d_in[0];   // lost_unit_id_seqs [B,S] int32
    const int*   lens = (const int*)d_in[1];   // lost_lengths [B] int32
    const float* pw   = (const float*)d_in[2]; // p_weights [B,S,L] float32
    // d_in[3] = max_word_length scalar (compile-time L_ = 20)
    float* out = (float*)d_out;
    int*   meta = (int*)d_ws;                  // 130 ints used

    hipLaunchKernelGGL(span_setup,   dim3(1),              dim3(B_),  0, stream, lens, meta);
    hipLaunchKernelGGL(span_cand,    dim3(N_ / (256 * 4)), dim3(256), 0, stream, lens, out);
    hipLaunchKernelGGL(span_compact, dim3((B_ * S_) / 8),  dim3(256), 0, stream, seqs, lens, pw, meta, out);
    hipLaunchKernelGGL(span_tail,    dim3(N_ / 256),       dim3(256), 0, stream, seqs, pw, meta, out);
}